// CheiralityLayer_67001489817747
// MI455X (gfx1250) — compile-verified
//
#include <hip/hip_runtime.h>
#include <math.h>
#include <stdint.h>

// Problem constants (from reference): B=16, C=6, H=480, W=640
#define BATCH 16
#define HH    480
#define WW    640
#define HWSZ  (HH * WW)                 // 307200
#define ROWS  8                         // compute rows per block
#define LROWS (ROWS + 2)                // + halo rows (y-stencil)
#define TPB   256                       // 8 wave32s
#define TILES_PER_IMG (HH / ROWS)       // 60
#define NBLK  (BATCH * TILES_PER_IMG)   // 960
#define CHUNKS_PER_ROW (WW * 4 / 16)    // 160 x 16B chunks per row
#define NCHUNK (LROWS * CHUNKS_PER_ROW) // 1600 chunks per channel tile

// ---- CDNA5 async DMA: global -> LDS, 16B per active lane, ASYNCcnt tracked ----
__device__ __forceinline__ void async_load_b128(unsigned lds_off, unsigned goff,
                                                const float* sbase) {
  asm volatile("global_load_async_to_lds_b128 %0, %1, %2"
               :: "v"(lds_off), "v"(goff), "s"(sbase)
               : "memory");
}
__device__ __forceinline__ void wait_asynccnt0() {
  asm volatile("s_wait_asynccnt 0x0" ::: "memory");
}

__global__ __launch_bounds__(TPB)
void cheirality_main(const float* __restrict__ img,
                     const float* __restrict__ pose,
                     const float* __restrict__ nf,
                     float* __restrict__ ws) {
  // s_ch0 is reused as the gray tile after phase 2 (in-place, race-free):
  // total LDS ~77 KB -> 4 workgroups per WGP (320 KB pool).
  __shared__ float s_ch0[LROWS * WW];   // 25600 B, becomes gray
  __shared__ float s_ch1[LROWS * WW];
  __shared__ float s_ch2[LROWS * WW];
  __shared__ float s_red[TPB / 32];

  const int tid = threadIdx.x;
  const int blk = blockIdx.x;
  const int b   = blk / TILES_PER_IMG;
  const int t   = blk - b * TILES_PER_IMG;
  const int r0  = t * ROWS;

  const float* p0 = img + ((size_t)b * 6 + 0) * HWSZ;
  const float* p1 = img + ((size_t)b * 6 + 1) * HWSZ;
  const float* p2 = img + ((size_t)b * 6 + 2) * HWSZ;

  const unsigned l0 = (unsigned)(uintptr_t)(void*)s_ch0;
  const unsigned l1 = (unsigned)(uintptr_t)(void*)s_ch1;
  const unsigned l2 = (unsigned)(uintptr_t)(void*)s_ch2;

  // ---- Phase 1: DMA three channel tiles (with clamped halo rows) into LDS ----
  for (int i = tid; i < NCHUNK; i += TPB) {
    int lr = i / CHUNKS_PER_ROW;
    int cc = i - lr * CHUNKS_PER_ROW;
    int gr = r0 - 1 + lr;
    gr = (gr < 0) ? 0 : ((gr > HH - 1) ? (HH - 1) : gr);
    unsigned goff = (unsigned)(gr * (WW * 4) + cc * 16);
    unsigned loff = (unsigned)(i * 16);
    async_load_b128(l0 + loff, goff, p0);
    async_load_b128(l1 + loff, goff, p1);
    async_load_b128(l2 + loff, goff, p2);
  }
  wait_asynccnt0();
  __syncthreads();

  // ---- Phase 2: gray = mean of 3 channels, written in place over s_ch0 ----
  for (int i = tid; i < LROWS * WW; i += TPB)
    s_ch0[i] = (s_ch0[i] + s_ch1[i] + s_ch2[i]) * (1.0f / 3.0f);
  __syncthreads();
  const float* s_gray = s_ch0;

  // ---- Phase 3: stencil + analytic contraction + exact GELU, accumulate ----
  const float Vx = pose[b * 6 + 0], Vy = pose[b * 6 + 1], Vz = pose[b * 6 + 2];
  const float Wx = pose[b * 6 + 3], Wy = pose[b * 6 + 4], Wz = pose[b * 6 + 5];

  const float* nfp0 = nf + ((size_t)b * 2 + 0) * HWSZ;
  const float* nfp1 = nf + ((size_t)b * 2 + 1) * HWSZ;

  float acc = 0.0f;
  const int QPR = WW / 4;  // 160 quads per row
  for (int q = tid; q < ROWS * QPR; q += TPB) {
    int lr = q / QPR;
    int c4 = (q - lr * QPR) * 4;
    int gr = r0 + lr;
    int sl = lr + 1;

    const float4 ctr = *(const float4*)(s_gray + sl * WW + c4);
    float left  = (c4 == 0)       ? 0.0f : s_gray[sl * WW + c4 - 1];
    float right = (c4 + 4 >= WW)  ? 0.0f : s_gray[sl * WW + c4 + 4];
    float4 up = *(const float4*)(s_gray + (sl - 1) * WW + c4);
    float4 dn = *(const float4*)(s_gray + (sl + 1) * WW + c4);
    if (gr == 0)      { up.x = up.y = up.z = up.w = 0.0f; }
    if (gr == HH - 1) { dn.x = dn.y = dn.z = dn.w = 0.0f; }

    const float4 f0 = *(const float4*)(nfp0 + (size_t)gr * WW + c4);
    const float4 f1 = *(const float4*)(nfp1 + (size_t)gr * WW + c4);

    float gx[4]  = { ctr.y - left, ctr.z - ctr.x, ctr.w - ctr.y, right - ctr.z };
    float gy[4]  = { dn.x - up.x, dn.y - up.y, dn.z - up.z, dn.w - up.w };
    float nfs[4] = { f0.x + f1.x, f0.y + f1.y, f0.z + f1.z, f0.w + f1.w };

    const float y = (float)gr;
    const float AV1  = y * Vz - Vy;                // row-invariant pieces
    const float yWz  = y * Wz;
    const float BW1c = (y * y + 1.0f) * Wx;
#pragma unroll
    for (int j = 0; j < 4; ++j) {
      float x   = (float)(c4 + j);
      float inv = rsqrtf(gx[j] * gx[j] + gy[j] * gy[j] + 1e-8f);
      float g0 = gx[j] * inv, g1 = gy[j] * inv;
      float AV0 = x * Vz - Vx;
      float BW0 = x * y * Wx - (x * x + 1.0f) * Wy + yWz;
      float BW1 = BW1c - x * y * Wy - x * Wz;
      float a   = g0 * AV0 + g1 * AV1;
      float bb  = nfs[j] - (g0 * BW0 + g1 * BW1);
      float tn  = -(a * bb);                       // -rho
      acc += 0.5f * tn * (1.0f + erff(tn * 0.70710678118654752f));
    }
  }

  // ---- deterministic block reduction (wave32 shuffles + LDS) ----
  for (int off = 16; off > 0; off >>= 1)
    acc += __shfl_down(acc, off, 32);
  if ((tid & 31) == 0) s_red[tid >> 5] = acc;
  __syncthreads();
  if (tid == 0) {
    float s = 0.0f;
#pragma unroll
    for (int i = 0; i < TPB / 32; ++i) s += s_red[i];
    ws[blk] = s;
  }
}

__global__ __launch_bounds__(TPB)
void cheirality_reduce(const float* __restrict__ ws, float* __restrict__ out) {
  __shared__ float sm[TPB];
  float s = 0.0f;
  for (int i = threadIdx.x; i < NBLK; i += TPB) s += ws[i];
  sm[threadIdx.x] = s;
  __syncthreads();
  for (int st = TPB / 2; st > 0; st >>= 1) {
    if (threadIdx.x < st) sm[threadIdx.x] += sm[threadIdx.x + st];
    __syncthreads();
  }
  if (threadIdx.x == 0)
    out[0] = sm[0] * (1.0f / (float)((size_t)BATCH * HWSZ));
}

extern "C" void kernel_launch(void* const* d_in, const int* in_sizes, int n_in,
                              void* d_out, int out_size, void* d_ws, size_t ws_size,
                              hipStream_t stream) {
  (void)in_sizes; (void)n_in; (void)out_size; (void)ws_size;
  const float* img  = (const float*)d_in[0];   // (16,6,480,640) f32
  const float* pose = (const float*)d_in[1];   // (16,6) f32
  const float* nf   = (const float*)d_in[2];   // (16,2,480,640) f32
  float* out = (float*)d_out;                  // scalar f32
  float* ws  = (float*)d_ws;                   // >= 960 floats of scratch

  cheirality_main<<<NBLK, TPB, 0, stream>>>(img, pose, nf, ws);
  cheirality_reduce<<<1, TPB, 0, stream>>>(ws, out);
}